// LQR_41120016892822
// MI455X (gfx1250) — compile-verified
//
#include <hip/hip_runtime.h>

// LQR on MI455X (gfx1250): one workgroup per batch element, V persisted in
// 320KB LDS across the sequential T=128 Riccati scan; all GEMMs via fp32 WMMA
// V_WMMA_F32_16X16X4_F32 with 2x2 register tile blocking (2 ds_load per wmma).

typedef __attribute__((ext_vector_type(2))) float v2f;
typedef __attribute__((ext_vector_type(8))) float v8f;

#define NTHREADS 512
#define NWAVES   16

#define TT   128
#define NN   128
#define MM   64
#define NTAUD 192

// LDS leading dims: dim+4 -> 16B-aligned rows, column-read stride = 4 banks
// (16 lanes span all 64 banks, conflict-free).
#define LDV  132
#define LDF  196
#define LDW  132
#define LDXU 68
#define LDUU 68

// float offsets into shared block
#define OFF_V   0                      // 128 x 132 = 16896
#define OFF_F   16896                  // 128 x 196 = 25088
#define OFF_W   41984                  // 128 x 132 = 16896 (VF / K|k buffer)
#define OFF_QXU 58880                  // 128 x 68  = 8704
#define OFF_QUU 67584                  // 64  x 68  = 4352
#define OFF_VEC 71936                  // vectors (640)
#define SM_FLOATS 72576                // 290304 bytes < 320KB

__device__ __forceinline__ v8f wmma4(v2f a, v2f b, v8f c) {
  // D = A(16x4,f32) * B(4x16,f32) + C(16x16,f32)
  return __builtin_amdgcn_wmma_f32_16x16x4_f32(false, a, false, b, (short)0, c,
                                               false, false);
}

// A fragment (rows i0..i0+15, cols k0..k0+3) from row-major A
__device__ __forceinline__ v2f ldA(const float* A, int lda, int i0, int k0, int lane) {
  int m = lane & 15;
  int kk = k0 + ((lane >> 4) << 1);
  const float* p = A + (i0 + m) * lda + kk;
  v2f r; r.x = p[0]; r.y = p[1];
  return r;
}
// A^T fragment: op(A)[m][k] = Asrc[k][i0+m]  (Asrc row-major, K x M)
__device__ __forceinline__ v2f ldAT(const float* A, int lda, int i0, int k0, int lane) {
  int m = lane & 15;
  int kk = k0 + ((lane >> 4) << 1);
  v2f r; r.x = A[kk * lda + i0 + m]; r.y = A[(kk + 1) * lda + i0 + m];
  return r;
}
// B fragment (rows k0..k0+3, cols j0..j0+15) from row-major B
__device__ __forceinline__ v2f ldB(const float* B, int ldb, int k0, int j0, int lane) {
  int n = lane & 15;
  int kk = k0 + ((lane >> 4) << 1);
  const float* p = B + kk * ldb + j0 + n;
  v2f r; r.x = p[0]; r.y = p[ldb];
  return r;
}
// C/D 16x16 f32 tile: VGPR r -> M = r (lanes 0-15) / r+8 (lanes 16-31), N = lane&15
__device__ __forceinline__ v8f ldC(const float* C, long ldc, int i0, int j0, int lane) {
  int n = lane & 15;
  int rb = i0 + ((lane >> 4) << 3);
  v8f c;
#pragma unroll
  for (int r = 0; r < 8; ++r) c[r] = C[(long)(rb + r) * ldc + j0 + n];
  return c;
}
__device__ __forceinline__ void stC(float* D, int ldd, int i0, int j0, int lane, v8f c) {
  int n = lane & 15;
  int rb = i0 + ((lane >> 4) << 3);
#pragma unroll
  for (int r = 0; r < 8; ++r) D[(rb + r) * ldd + j0 + n] = c[r];
}

// Workgroup GEMM: D(MxN) = op(A)(MxK) * B(KxN) [+ Cin].
// Each wave owns a (16*BM x 16*BN) macro-tile: A/B fragments are shared across
// the register block, cutting LDS loads per wmma (2x2 -> 2 ds_load/wmma).
template <bool ATRANS, int BM, int BN>
__device__ void wgemm(const float* A, int lda, const float* B, int ldb,
                      const float* Cin, long ldc, float* D, int ldd,
                      int M, int N, int K, int wave, int lane) {
  const int mt = M / (16 * BM), nt = N / (16 * BN);
  for (int tile = wave; tile < mt * nt; tile += NWAVES) {
    const int i0 = (tile / nt) * (16 * BM);
    const int j0 = (tile % nt) * (16 * BN);
    v8f acc[BM][BN];
#pragma unroll
    for (int bm = 0; bm < BM; ++bm)
#pragma unroll
      for (int bn = 0; bn < BN; ++bn) {
        if (Cin) {
          acc[bm][bn] = ldC(Cin, ldc, i0 + 16 * bm, j0 + 16 * bn, lane);
        } else {
#pragma unroll
          for (int r = 0; r < 8; ++r) acc[bm][bn][r] = 0.f;
        }
      }
#pragma unroll 2
    for (int k0 = 0; k0 < K; k0 += 4) {
      v2f a[BM], b[BN];
#pragma unroll
      for (int bm = 0; bm < BM; ++bm)
        a[bm] = ATRANS ? ldAT(A, lda, i0 + 16 * bm, k0, lane)
                       : ldA(A, lda, i0 + 16 * bm, k0, lane);
#pragma unroll
      for (int bn = 0; bn < BN; ++bn) b[bn] = ldB(B, ldb, k0, j0 + 16 * bn, lane);
#pragma unroll
      for (int bm = 0; bm < BM; ++bm)
#pragma unroll
        for (int bn = 0; bn < BN; ++bn)
          acc[bm][bn] = wmma4(a[bm], b[bn], acc[bm][bn]);
    }
#pragma unroll
    for (int bm = 0; bm < BM; ++bm)
#pragma unroll
      for (int bn = 0; bn < BN; ++bn)
        stC(D, ldd, i0 + 16 * bm, j0 + 16 * bn, lane, acc[bm][bn]);
  }
}

__global__ __launch_bounds__(NTHREADS) void lqr_bwd(
    const float* __restrict__ Ag_, const float* __restrict__ Bg_,
    const float* __restrict__ cg_, const float* __restrict__ Qg_,
    const float* __restrict__ pg_, float* __restrict__ wsK) {
  __shared__ __align__(16) float sm[SM_FLOATS];
  float* sV   = sm + OFF_V;    // V (symmetric), later Vn
  float* sF   = sm + OFF_F;    // F = [A | B]
  float* sW   = sm + OFF_W;    // VF block, then K|k (64 x 129)
  float* sQxu = sm + OFF_QXU;
  float* sQuu = sm + OFF_QUU;  // Quu, then its Cholesky L (lower)
  float* vec  = sm + OFF_VEC;
  float* swv = vec;            // 128 : w = V c + v
  float* sqx = vec + 128;      // 128
  float* squ = vec + 256;      // 64
  float* scv = vec + 320;      // 128
  float* svv = vec + 448;      // 128 : v

  const int tid = threadIdx.x, lane = tid & 31, wave = tid >> 5;
  const int b = blockIdx.x;

  for (int i = tid; i < 128 * LDV; i += NTHREADS) sV[i] = 0.f;
  if (tid < 128) svv[tid] = 0.f;
  __syncthreads();

  for (int t = TT - 1; t >= 0; --t) {
    const long bt = (long)b * TT + t;
    const float* Ag = Ag_ + bt * (NN * NN);
    const float* Bg = Bg_ + bt * (NN * MM);
    const float* Qg = Qg_ + bt * (NTAUD * NTAUD);
    const float* pg = pg_ + bt * NTAUD;
    const float* cg = cg_ + bt * NN;

    // ---- load F = [A | B] into LDS (float4), plus c ----
    const float4* A4 = (const float4*)Ag;
    for (int idx = tid; idx < 128 * 32; idx += NTHREADS) {
      int m = idx >> 5, q = idx & 31;
      *(float4*)&sF[m * LDF + (q << 2)] = A4[idx];
    }
    const float4* B4 = (const float4*)Bg;
    for (int idx = tid; idx < 128 * 16; idx += NTHREADS) {
      int m = idx >> 4, q = idx & 15;
      *(float4*)&sF[m * LDF + 128 + (q << 2)] = B4[idx];
    }
    if (tid < 128) scv[tid] = cg[tid];

    // prefetch next timestep's Q/A/B into cache while we compute
    if (t > 0) {
      const char* nq = (const char*)(Qg_ + (bt - 1) * (NTAUD * NTAUD));
      const char* na = (const char*)(Ag_ + (bt - 1) * (NN * NN));
      const char* nb = (const char*)(Bg_ + (bt - 1) * (NN * MM));
      for (int l = tid; l < 576; l += NTHREADS) __builtin_prefetch(nq + l * 256, 0, 1);
      for (int l = tid; l < 256; l += NTHREADS) __builtin_prefetch(na + l * 256, 0, 1);
      for (int l = tid; l < 128; l += NTHREADS) __builtin_prefetch(nb + l * 256, 0, 1);
    }
    __syncthreads();

    // ---- w = V c + v ----
    if (tid < 128) {
      float s = svv[tid];
      const float* vr = sV + tid * LDV;
      for (int m = 0; m < 128; ++m) s += vr[m] * scv[m];
      swv[tid] = s;
    }
    __syncthreads();

    // ---- qhat = p + F^T w ----
    if (tid < 192) {
      float s = pg[tid];
      for (int n = 0; n < 128; ++n) s += sF[n * LDF + tid] * swv[n];
      if (tid < 128) sqx[tid] = s; else squ[tid - 128] = s;
    }
    __syncthreads();

    // ---- VFu = V * F[:,128:192]  (128x64) -> sW ----
    wgemm<false, 2, 1>(sV, LDV, sF + 128, LDF, nullptr, 0, sW, LDW,
                       128, 64, 128, wave, lane);
    __syncthreads();
    // ---- Qxu = Q[0:128,128:] + Fx^T VFu ; Quu = Q[128:,128:] + Fu^T VFu ----
    wgemm<true, 2, 1>(sF, LDF, sW, LDW, Qg + 128, NTAUD, sQxu, LDXU,
                      128, 64, 128, wave, lane);
    wgemm<true, 1, 1>(sF + 128, LDF, sW, LDW, Qg + 128 * NTAUD + 128, NTAUD,
                      sQuu, LDUU, 64, 64, 128, wave, lane);
    __syncthreads();
    // ---- VFx = V * F[:,0:128]  (128x128) -> sW ----
    wgemm<false, 2, 2>(sV, LDV, sF, LDF, nullptr, 0, sW, LDW,
                       128, 128, 128, wave, lane);
    __syncthreads();
    // ---- Qxx = Q[0:128,0:128] + Fx^T VFx -> overwrite sV (V is dead now) ----
    wgemm<true, 2, 2>(sF, LDF, sW, LDW, Qg, NTAUD, sV, LDV,
                      128, 128, 128, wave, lane);
    __syncthreads();

    // ---- Cholesky: Quu = L L^T (in-place, lower) ----
    for (int kc = 0; kc < 64; ++kc) {
      if (tid == 0) sQuu[kc * LDUU + kc] = sqrtf(sQuu[kc * LDUU + kc]);
      __syncthreads();
      const float dk = sQuu[kc * LDUU + kc];
      const int rem = 63 - kc;
      if (tid < rem) sQuu[(kc + 1 + tid) * LDUU + kc] /= dk;
      __syncthreads();
      for (int e = tid; e < rem * rem; e += NTHREADS) {
        int i = kc + 1 + e / rem, j = kc + 1 + e % rem;
        if (j <= i) sQuu[i * LDUU + j] -= sQuu[i * LDUU + kc] * sQuu[j * LDUU + kc];
      }
      __syncthreads();
    }

    // ---- [K | k] = -Quu^{-1} [Qxu^T | qu]  -> sW rows 0..63, cols 0..128 ----
    if (tid < 129) {
      const int c = tid;
      for (int u = 0; u < 64; ++u) {            // L y = rhs
        float s = (c < 128) ? sQxu[c * LDXU + u] : squ[u];
        for (int j = 0; j < u; ++j) s -= sQuu[u * LDUU + j] * sW[j * LDW + c];
        sW[u * LDW + c] = s / sQuu[u * LDUU + u];
      }
      for (int u = 63; u >= 0; --u) {           // L^T x = y, store -x
        float s = -sW[u * LDW + c];
        for (int j = u + 1; j < 64; ++j) s -= sQuu[j * LDUU + u] * sW[j * LDW + c];
        sW[u * LDW + c] = s / sQuu[u * LDUU + u];
      }
    }
    __syncthreads();

    // ---- spill K|k for the forward pass ----
    float* wk = wsK + bt * (64 * 129);
    for (int idx = tid; idx < 64 * 129; idx += NTHREADS) {
      int u = idx / 129, c = idx - u * 129;
      wk[idx] = sW[u * LDW + c];
    }

    // ---- Vn = Qxx + Qxu K   (since Quu K = -Qux, cross terms cancel) ----
    wgemm<false, 2, 2>(sQxu, LDXU, sW, LDW, sV, (long)LDV, sV, LDV,
                       128, 128, 64, wave, lane);
    // ---- vn = qx + Qxu k ----
    if (tid < 128) {
      float s = sqx[tid];
      for (int u = 0; u < 64; ++u) s += sQxu[tid * LDXU + u] * sW[u * LDW + 128];
      svv[tid] = s;
    }
    __syncthreads();
  }
}

__global__ __launch_bounds__(512) void lqr_fwd(
    const float* __restrict__ Ag_, const float* __restrict__ Bg_,
    const float* __restrict__ cg_, const float* __restrict__ xinit,
    const float* __restrict__ wsK, float* __restrict__ out) {
  __shared__ float sx[128], sxn[128], su[64], sp[512];
  const int tid = threadIdx.x;
  const int b = blockIdx.x;
  if (tid < 128) sx[tid] = xinit[b * 128 + tid];
  __syncthreads();
  for (int t = 0; t < TT; ++t) {
    const long bt = (long)b * TT + t;
    const float* Kk = wsK + bt * (64 * 129);
    {  // u = K x + k  (64 rows x 8 partials)
      int r = tid >> 3, pp = tid & 7; float s = 0.f;
      for (int jj = 0; jj < 16; ++jj) { int j = pp + (jj << 3); s += Kk[r * 129 + j] * sx[j]; }
      sp[tid] = s;
    }
    __syncthreads();
    if (tid < 64) {
      float s = Kk[tid * 129 + 128];
      for (int pp = 0; pp < 8; ++pp) s += sp[(tid << 3) + pp];
      su[tid] = s;
    }
    __syncthreads();
    if (tid < 192) out[bt * 192 + tid] = (tid < 128) ? sx[tid] : su[tid - 128];
    const float* Ag = Ag_ + bt * (128 * 128);
    const float* Bg = Bg_ + bt * (128 * 64);
    const float* cg = cg_ + bt * 128;
    {  // xn = A x + B u + c  (128 rows x 4 partials)
      int r = tid >> 2, pp = tid & 3; float s = 0.f;
      for (int jj = 0; jj < 32; ++jj) { int j = pp + (jj << 2); s += Ag[r * 128 + j] * sx[j]; }
      for (int jj = 0; jj < 16; ++jj) { int j = pp + (jj << 2); s += Bg[r * 64 + j] * su[j]; }
      sp[tid] = s;
    }
    __syncthreads();
    if (tid < 128) {
      float s = cg[tid];
      for (int pp = 0; pp < 4; ++pp) s += sp[(tid << 2) + pp];
      sxn[tid] = s;
    }
    __syncthreads();
    if (tid < 128) sx[tid] = sxn[tid];
    __syncthreads();
  }
}

extern "C" void kernel_launch(void* const* d_in, const int* in_sizes, int n_in,
                              void* d_out, int out_size, void* d_ws, size_t ws_size,
                              hipStream_t stream) {
  (void)in_sizes; (void)n_in; (void)out_size; (void)ws_size;
  const float* A  = (const float*)d_in[0];
  const float* Bm = (const float*)d_in[1];
  const float* c1 = (const float*)d_in[2];
  const float* Q  = (const float*)d_in[3];
  const float* p  = (const float*)d_in[4];
  const float* xi = (const float*)d_in[5];
  float* out = (float*)d_out;
  float* wsK = (float*)d_ws;  // needs 32*128*64*129*4 = ~129 MB

  lqr_bwd<<<32, NTHREADS, 0, stream>>>(A, Bm, c1, Q, p, wsK);
  lqr_fwd<<<32, 512, 0, stream>>>(A, Bm, c1, xi, wsK, out);
}